// RelationProposalNetwork_2388001817049
// MI455X (gfx1250) — compile-verified
//
#include <hip/hip_runtime.h>

typedef __attribute__((ext_vector_type(2))) float v2f;
typedef __attribute__((ext_vector_type(8))) float v8f;

#define NBOX 512
#define CDIM 256
#define HDIM 256
#define INDIM 516   // 2*C + 4

// ---------------------------------------------------------------------------
// Kernel 1: Vrow[i,h] = feats[i,:] . W1[h, 0:256]   + b1[h]
//           Vcol[i,h] = feats[i,:] . W1[h, 256:512]
// One wave computes one 16(i) x 16(h) tile via V_WMMA_F32_16X16X4_F32,
// stepping K=256 in chunks of 4. 1024 waves total.
// A frag (16x4 f32): lane<16 -> K={0,1}; lane>=16 -> K={2,3}; M = lane&15.
// B frag (4x16 f32): mirror layout, N = lane&15.
// ---------------------------------------------------------------------------
__global__ __launch_bounds__(256) void rpn_rowcol_kernel(
    const float* __restrict__ feats, const float* __restrict__ W1,
    const float* __restrict__ b1, float* __restrict__ Vrow,
    float* __restrict__ Vcol) {
  const int lane  = threadIdx.x & 31;
  const int wave  = (blockIdx.x * blockDim.x + threadIdx.x) >> 5;
  const int which = wave & 1;          // 0 = row (W1a), 1 = col (W1b)
  const int ht    = (wave >> 1) & 15;  // h tile [0,16)
  const int it    = wave >> 5;         // i tile [0,32)
  const int mrow  = lane & 15;
  const int kb    = (lane >> 4) << 1;  // 0 or 2
  const int hi8   = (lane >> 4) << 3;  // 0 or 8

  const int i0 = it * 16;
  const int h0 = ht * 16;
  const float* arow = feats + (i0 + mrow) * CDIM + kb;
  const float* brow = W1 + (h0 + mrow) * INDIM + which * CDIM + kb;

  v8f acc = {};
  #pragma unroll 8
  for (int k = 0; k < CDIM; k += 4) {
    v2f a = *(const v2f*)(arow + k);   // 8B aligned: k, kb even
    v2f b = *(const v2f*)(brow + k);
    acc = __builtin_amdgcn_wmma_f32_16x16x4_f32(
        false, a, false, b, (short)0, acc, false, false);
  }

  const float bias = (which == 0) ? b1[h0 + mrow] : 0.0f;
  float* dst = (which == 0) ? Vrow : Vcol;
  #pragma unroll
  for (int v = 0; v < 8; ++v) {
    // element (M = v + hi8, N = mrow) of the 16x16 C tile
    dst[(i0 + v + hi8) * HDIM + h0 + mrow] = acc[v] + bias;
  }
}

// ---------------------------------------------------------------------------
// Kernel 2: fused pairwise scoring.
// wave -> (ib = 4 consecutive i's, jt = 16 consecutive j's), all 256 h.
// For each h-group of 16:
//   C_in[M,N] = Vrow[i, h0+N] + Vcol[j0+M, h0+N]           (bias folded in Vrow)
//   D = geom(16x4) x W1g^T(4x16) + C_in                    (one WMMA)
//   acc[M] += W2[h0+N] * max(D[M,N], 0)                    (VALU)
// Then reduce acc over N (16 lanes per half) with shfl_xor and store.
// 4096 waves total; intermediate (512,512,256) tensor lives only in VGPRs.
// ---------------------------------------------------------------------------
__global__ __launch_bounds__(256) void rpn_pair_kernel(
    const float* __restrict__ boxes, const float* __restrict__ W1,
    const float* __restrict__ Vrow, const float* __restrict__ Vcol,
    const float* __restrict__ W2, const float* __restrict__ b2,
    float* __restrict__ out) {
  const int lane = threadIdx.x & 31;
  const int wave = (blockIdx.x * blockDim.x + threadIdx.x) >> 5;
  const int jt   = wave & 31;          // j tile [0,32)
  const int ib   = (wave >> 5) << 2;   // base i (4 i's per wave)
  const int mrow = lane & 15;          // M (=j offset) for A, N (=h offset) for B/C
  const int kb   = (lane >> 4) << 1;   // geom K base: 0 or 2
  const int hi8  = (lane >> 4) << 3;   // 0 or 8
  const int j0   = jt * 16;

  // A fragments: geom[i, j0+M, kb..kb+1] = |boxes[i,g] - boxes[j,g]|
  const v2f bj = *(const v2f*)(boxes + (j0 + mrow) * 4 + kb);
  v2f afrag[4];
  #pragma unroll
  for (int ii = 0; ii < 4; ++ii) {
    v2f bi = *(const v2f*)(boxes + (ib + ii) * 4 + kb);
    afrag[ii].x = fabsf(bi.x - bj.x);
    afrag[ii].y = fabsf(bi.y - bj.y);
  }

  const float* w1g  = W1 + mrow * INDIM + 2 * CDIM + kb;  // + h0*INDIM per group
  const float* colp = Vcol + (j0 + hi8) * HDIM + mrow;    // + h0 per group
  v8f acc[4] = {{}, {}, {}, {}};

  for (int hg = 0; hg < 16; ++hg) {
    const int h0 = hg * 16;
    const v2f b = *(const v2f*)(w1g + h0 * INDIM);  // W1g[h0+mrow, kb..kb+1]
    const float w2h = W2[h0 + mrow];
    float col[8];
    #pragma unroll
    for (int v = 0; v < 8; ++v) col[v] = colp[v * HDIM + h0];
    float r[4];
    #pragma unroll
    for (int ii = 0; ii < 4; ++ii) r[ii] = Vrow[(ib + ii) * HDIM + h0 + mrow];

    #pragma unroll
    for (int ii = 0; ii < 4; ++ii) {
      v8f cin;
      #pragma unroll
      for (int v = 0; v < 8; ++v) cin[v] = r[ii] + col[v];
      v8f d = __builtin_amdgcn_wmma_f32_16x16x4_f32(
          false, afrag[ii], false, b, (short)0, cin, false, false);
      #pragma unroll
      for (int v = 0; v < 8; ++v)
        acc[ii][v] = fmaf(fmaxf(d[v], 0.0f), w2h, acc[ii][v]);
    }
  }

  // Reduce over h (N) within each 16-lane half; lane 0 / lane 16 keep results.
  const float b2v = b2[0];
  #pragma unroll
  for (int ii = 0; ii < 4; ++ii) {
    #pragma unroll
    for (int v = 0; v < 8; ++v) {
      float s = acc[ii][v];
      s += __shfl_xor(s, 1, 32);
      s += __shfl_xor(s, 2, 32);
      s += __shfl_xor(s, 4, 32);
      s += __shfl_xor(s, 8, 32);
      if (mrow == 0)
        out[(ib + ii) * NBOX + j0 + v + hi8] = s + b2v;
    }
  }
}

// ---------------------------------------------------------------------------
// Launch: inputs = feats(512x256), boxes(512x4), W1(256x516), b1(256),
//         W2(1x256), b2(1); out = scores(512x512) f32.
// Workspace: Vrow (512*256 f32) | Vcol (512*256 f32) = 1 MB.
// ---------------------------------------------------------------------------
extern "C" void kernel_launch(void* const* d_in, const int* in_sizes, int n_in,
                              void* d_out, int out_size, void* d_ws, size_t ws_size,
                              hipStream_t stream) {
  const float* feats = (const float*)d_in[0];
  const float* boxes = (const float*)d_in[1];
  const float* W1    = (const float*)d_in[2];
  const float* b1    = (const float*)d_in[3];
  const float* W2    = (const float*)d_in[4];
  const float* b2    = (const float*)d_in[5];
  float* out  = (float*)d_out;
  float* Vrow = (float*)d_ws;
  float* Vcol = Vrow + NBOX * HDIM;

  // 1024 waves: 32 i-tiles x 16 h-tiles x {row,col}
  rpn_rowcol_kernel<<<128, 256, 0, stream>>>(feats, W1, b1, Vrow, Vcol);
  // 4096 waves: 128 i-quads x 32 j-tiles
  rpn_pair_kernel<<<512, 256, 0, stream>>>(boxes, W1, Vrow, Vcol, W2, b2, out);
}